// ResidualBlock_65532611002532
// MI455X (gfx1250) — compile-verified
//
#include <hip/hip_runtime.h>

typedef float v2f __attribute__((ext_vector_type(2)));
typedef float v8f __attribute__((ext_vector_type(8)));

__device__ __forceinline__ float leaky(float x) {
    return x > 0.0f ? x : 0.2f * x;
}

__device__ __forceinline__ v8f wmma_f32_k4(v2f a, v2f b, v8f c) {
    // D(16x16,f32) = A(16x4,f32) * B(4x16,f32) + C
    return __builtin_amdgcn_wmma_f32_16x16x4_f32(
        /*neg_a=*/false, a, /*neg_b=*/false, b,
        /*c_mod=*/(short)0, c, /*reuse_a=*/false, /*reuse_b=*/false);
}

// ---------------------------------------------------------------------------
// Generic fused GEMM: out[M,N] = leaky(X[M,K] @ W[N,K]^T + bias[N]) (+ add)
// One wave per 16x16 output tile; fp32 WMMA, K stepped by 4.
// A layout (16x4 f32): lane<16 -> row=lane, cols k,k+1 ; lane>=16 -> row=lane-16, cols k+2,k+3
// B layout (4x16 f32): lane<16 -> col=lane, rows k,k+1 ; lane>=16 -> col=lane-16, rows k+2,k+3
// ---------------------------------------------------------------------------
__global__ __launch_bounds__(256) void gemm_leaky_wmma(
    const float* __restrict__ X, const float* __restrict__ W,
    const float* __restrict__ bias, const float* add,
    float* out, int M, int N, int K)
{
    const int lane = threadIdx.x & 31;
    const int wv   = threadIdx.x >> 5;
    const int nTilesN = N >> 4;
    const int nTiles  = (M >> 4) * nTilesN;
    const int tile = blockIdx.x * 8 + wv;   // uniform per wave
    if (tile >= nTiles) return;             // whole-wave exit (EXEC stays all-ones)

    const int m0 = (tile / nTilesN) << 4;
    const int n0 = (tile % nTilesN) << 4;
    const int lm = lane & 15;
    const int lh = lane >> 4;

    const float* arow = X + (size_t)(m0 + lm) * (size_t)K + (lh << 1);
    const float* brow = W + (size_t)(n0 + lm) * (size_t)K + (lh << 1);

    v8f acc = {};
    #pragma unroll 4
    for (int kk = 0; kk < K; kk += 4) {
        v2f a = *reinterpret_cast<const v2f*>(arow + kk);
        v2f b = *reinterpret_cast<const v2f*>(brow + kk);
        acc = wmma_f32_k4(a, b, acc);
    }

    // C/D layout: VGPR r holds (row = m0 + r + 8*lh, col = n0 + lm)
    const float bcol = bias[n0 + lm];
    float*       orow = out + (size_t)(m0 + (lh << 3)) * (size_t)N + (n0 + lm);
    const float* airow = add ? add + (size_t)(m0 + (lh << 3)) * (size_t)N + (n0 + lm)
                             : (const float*)nullptr;
    #pragma unroll
    for (int r = 0; r < 8; ++r) {
        float v = leaky(acc[r] + bcol);
        if (airow) v += airow[(size_t)r * N];
        orow[(size_t)r * N] = v;
    }
}

// ---------------------------------------------------------------------------
// Local Feature Aggregation: one wave per point.
//   geo(k) = [rel.xyz, |rel|]  (k = 0..31 neighbors, one per lane)
//   mlp    = leaky(geo @ W^T + b)          (C channels, WMMA 16x16x4 f32)
//   out[p] = concat(mean_k mlp, mean_k gathered_feats)   (2*C channels, FD==C)
// ---------------------------------------------------------------------------
template <int FD, int C>
__global__ __launch_bounds__(128) void lfa_wmma(
    const float* __restrict__ points, const float* __restrict__ feats,
    const int* __restrict__ idx, const float* __restrict__ W,
    const float* __restrict__ bias, float* __restrict__ out,
    int nper, int P)
{
    constexpr int KNN   = 32;
    constexpr int WAVES = 4;
    constexpr int CT    = C / 16;

    __shared__ __align__(16) float s_geo[WAVES * KNN * 4];
    __shared__ __align__(16) float s_ft [WAVES * KNN * FD];

    const int lane = threadIdx.x & 31;
    const int wv   = threadIdx.x >> 5;
    const int p    = blockIdx.x * WAVES + wv;      // uniform per wave
    const int pp   = (p < P) ? p : (P - 1);        // clamp (keep barriers uniform)

    const int       b    = pp / nper;
    const long long prow = (long long)pp;

    // center point (wave-uniform -> scalarized loads)
    const float cx = points[prow * 3 + 0];
    const float cy = points[prow * 3 + 1];
    const float cz = points[prow * 3 + 2];

    // per-lane neighbor gather (random, serviced by 192MB L2)
    const int       nb   = idx[prow * KNN + lane];
    const long long nrow = (long long)b * nper + nb;
    const float nx = points[nrow * 3 + 0];
    const float ny = points[nrow * 3 + 1];
    const float nz = points[nrow * 3 + 2];
    const float rx = cx - nx, ry = cy - ny, rz = cz - nz;
    const float dd = sqrtf(rx * rx + ry * ry + rz * rz);

    // stage geo rows in LDS (row = neighbor, 4 floats)
    float* g = s_geo + wv * KNN * 4;
    g[lane * 4 + 0] = rx;
    g[lane * 4 + 1] = ry;
    g[lane * 4 + 2] = rz;
    g[lane * 4 + 3] = dd;

    // stage gathered neighbor features in LDS (float4 vectorized)
    float* lf = s_ft + wv * KNN * FD;
    {
        const float4* fsrc = reinterpret_cast<const float4*>(feats + nrow * FD);
        float4*       ldst = reinterpret_cast<float4*>(lf + lane * FD);
        #pragma unroll
        for (int i = 0; i < FD / 4; ++i) ldst[i] = fsrc[i];
    }
    __syncthreads();

    const int lm = lane & 15;
    const int lh = lane >> 4;

    // WMMA MLP: two neighbor tiles (rows 0-15, 16-31) x CT channel tiles
    v8f acc[2][CT];
    #pragma unroll
    for (int t = 0; t < 2; ++t) {
        // A tile from LDS in native 16x4 layout
        v2f a = *reinterpret_cast<const v2f*>(g + (t * 16 + lm) * 4 + (lh << 1));
        #pragma unroll
        for (int ct = 0; ct < CT; ++ct) {
            v2f bb = *reinterpret_cast<const v2f*>(W + (ct * 16 + lm) * 4 + (lh << 1));
            v8f z = {};
            acc[t][ct] = wmma_f32_k4(a, bb, z);
        }
    }

    // bias + leaky per element, then mean over the 32 neighbors (rows of C)
    float mlpm[CT];
    #pragma unroll
    for (int ct = 0; ct < CT; ++ct) {
        const float bcol = bias[ct * 16 + lm];
        float s = 0.0f;
        #pragma unroll
        for (int r = 0; r < 8; ++r) s += leaky(acc[0][ct][r] + bcol);
        #pragma unroll
        for (int r = 0; r < 8; ++r) s += leaky(acc[1][ct][r] + bcol);
        s += __shfl_xor(s, 16, 32);           // combine row halves (M=r vs M=r+8)
        mlpm[ct] = s * (1.0f / 32.0f);        // now valid in both lane halves
    }

    // mean of gathered features: lane handles channel (lane % FD)
    const int ch = lane % FD;
    float fs = 0.0f;
    #pragma unroll
    for (int k = 0; k < KNN; ++k) fs += lf[k * FD + ch];
    fs *= (1.0f / 32.0f);

    if (p < P) {
        float* orow = out + (long long)p * (2 * C);
        if (lane < 16) {
            #pragma unroll
            for (int ct = 0; ct < CT; ++ct) orow[ct * 16 + lane] = mlpm[ct];
        }
        if (lane < C) orow[C + lane] = fs;    // FD == C, channel == lane
    }
}

// ---------------------------------------------------------------------------
// Host launcher
// ---------------------------------------------------------------------------
extern "C" void kernel_launch(void* const* d_in, const int* in_sizes, int n_in,
                              void* d_out, int out_size, void* d_ws, size_t ws_size,
                              hipStream_t stream) {
    (void)in_sizes; (void)n_in; (void)out_size; (void)ws_size;

    const float* points  = (const float*)d_in[0];
    const float* feats   = (const float*)d_in[1];
    const int*   knn     = (const int*)d_in[2];
    const float* w_init  = (const float*)d_in[3];
    const float* b_init  = (const float*)d_in[4];
    const float* w_lfa1  = (const float*)d_in[5];
    const float* b_lfa1  = (const float*)d_in[6];
    const float* w_lfa2  = (const float*)d_in[7];
    const float* b_lfa2  = (const float*)d_in[8];
    const float* w_final = (const float*)d_in[9];
    const float* b_final = (const float*)d_in[10];
    const float* w_res   = (const float*)d_in[11];
    const float* b_res   = (const float*)d_in[12];
    float* out = (float*)d_out;

    const int Nn = 65536;            // points per batch
    const int M  = 2 * Nn;           // 131072 total rows
    const int Dd = 64;

    float* ws  = (float*)d_ws;
    float* x16 = ws;                          // M*16 floats
    float* x32 = x16 + (size_t)M * 16;        // M*32 floats
    float* x64 = x32 + (size_t)M * 32;        // M*64 floats

    // 1) x16 = leaky(features @ w_init^T + b_init)
    {
        const int tiles = (M / 16) * (16 / 16);
        gemm_leaky_wmma<<<(tiles + 7) / 8, 256, 0, stream>>>(
            feats, w_init, b_init, nullptr, x16, M, 16, Dd);
    }
    // 2) residual = leaky(features @ w_res^T + b_res) -> d_out
    {
        const int tiles = (M / 16) * (64 / 16);
        gemm_leaky_wmma<<<(tiles + 7) / 8, 256, 0, stream>>>(
            feats, w_res, b_res, nullptr, out, M, 64, Dd);
    }
    // 3) x32 = LFA1(points, x16)
    lfa_wmma<16, 16><<<M / 4, 128, 0, stream>>>(
        points, x16, knn, w_lfa1, b_lfa1, x32, Nn, M);
    // 4) x64 = LFA2(points, x32)
    lfa_wmma<32, 32><<<M / 4, 128, 0, stream>>>(
        points, x32, knn, w_lfa2, b_lfa2, x64, Nn, M);
    // 5) d_out = leaky(x64 @ w_final^T + b_final) + residual(d_out), in place
    {
        const int tiles = (M / 16) * (64 / 16);
        gemm_leaky_wmma<<<(tiles + 7) / 8, 256, 0, stream>>>(
            x64, w_final, b_final, out, out, M, 64, Dd);
    }
}